// BlockELLLinear_84791244357907
// MI455X (gfx1250) — compile-verified
//
#include <hip/hip_runtime.h>

// Block-ELL sparse linear on gfx1250: y[n, r*16+o] = sum_k sum_i x[n, col[r,k]*16+i] * V[r,k,o,i] + bias
// fp32 WMMA (V_WMMA_F32_16X16X4_F32), wave32.
//   - one workgroup (1024 thr / 32 waves) per 16-row n-tile; x tile staged once in LDS
//     with +4-float row padding -> conflict-free ds_load_b64 gathers
//   - col_indices read as SCALAR loads (KMcnt path, grouped x4 -> s_load_b128) so index
//     waits never drain the vector-load pipeline
//   - values streamed from L2 (global b64 clauses), 4 accumulating WMMAs per (r,k)

typedef __attribute__((ext_vector_type(2))) float v2f;
typedef __attribute__((ext_vector_type(8))) float v8f;

#define BB 16             // block size
#define RR 256            // output block rows
#define KK 64             // nonzero blocks per row
#define NN 8192           // batch
#define XSTRIDE 4096      // floats per x / y row (C*B)
#define LPAD 4            // LDS row padding (floats)
#define LSTRIDE (XSTRIDE + LPAD)          // 4100 floats
#define LDS_BYTES (BB * LSTRIDE * 4)      // 262400 B < 320 KB WGP LDS
#define RPW 8             // block-rows r per wave (32 waves * 8 = 256)
#define KG 4              // k-group size (4 indices -> one s_load_b128)

__global__ __launch_bounds__(1024) void blockell_lds_wmma_f32(
    const float* __restrict__ x,       // (N, 4096)
    const float* __restrict__ values,  // (R, K, 16, 16)  [r][k][o][i]
    const float* __restrict__ bias,    // (4096)
    const int*   __restrict__ col,     // (R, K)
    float* __restrict__ out)           // (N, 4096)
{
    extern __shared__ float xtile[];   // [16][LSTRIDE]

    const int n0 = blockIdx.x * BB;

    // ---- Stage x tile: 16 rows x 4096 floats. Thread t fills row i, float4 column t.
    {
        const float* src = x + (size_t)n0 * XSTRIDE;
        const int t = threadIdx.x;     // 0..1023 == one full row per iteration
        #pragma unroll
        for (int i = 0; i < BB; ++i) {
            float4 v = ((const float4*)(src + (size_t)i * XSTRIDE))[t];
            *(float4*)(&xtile[i * LSTRIDE + t * 4]) = v;
        }
    }
    __syncthreads();

    const int lane = threadIdx.x & 31;
    const int lo   = lane & 15;        // A row / B column selector
    const int hi   = lane >> 4;        // half-wave selector
    const int wave = threadIdx.x >> 5; // 0..31

    // Per-lane LDS gather base: A[M=lo][i = c*16 + 4q + j + 2*hi], j in {0,1} contiguous
    const float* lbase = &xtile[lo * LSTRIDE + 2 * hi];

    #pragma unroll 1
    for (int j = 0; j < RPW; ++j) {
        // Scalarize r so col/values addressing is provably uniform -> SMEM loads (KMcnt).
        const int rs = __builtin_amdgcn_readfirstlane(wave * RPW + j);
        const int* colr = col + rs * KK;
        // B base: values[rs][k][o=lo][i = 4q + jj + 2*hi]
        const float* vb = values + (size_t)(rs * KK) * (BB * BB) + lo * BB + 2 * hi;

        v8f acc = {};

        #pragma unroll 1
        for (int kk = 0; kk < KK; kk += KG) {
            // 4 consecutive uniform indices: scalar loads through the constant cache.
            int cg[KG];
            #pragma unroll
            for (int u = 0; u < KG; ++u)
                cg[u] = __builtin_amdgcn_readfirstlane(colr[kk + u]);

            // prefetch next group's values blocks (emits global_prefetch_b8)
            __builtin_prefetch(vb + (size_t)((kk + KG) & (KK - 1)) * (BB * BB), 0, 1);

            #pragma unroll
            for (int u = 0; u < KG; ++u) {
                const float* vbk = vb + (size_t)(kk + u) * (BB * BB);
                const v2f b0 = *(const v2f*)(vbk + 0);
                const v2f b1 = *(const v2f*)(vbk + 4);
                const v2f b2 = *(const v2f*)(vbk + 8);
                const v2f b3 = *(const v2f*)(vbk + 12);

                // A gather from LDS (conflict-free: banks 4*lo + 4q + 2*hi)
                const float* la = lbase + cg[u] * BB;
                const v2f a0 = *(const v2f*)(la + 0);
                const v2f a1 = *(const v2f*)(la + 4);
                const v2f a2 = *(const v2f*)(la + 8);
                const v2f a3 = *(const v2f*)(la + 12);

                acc = __builtin_amdgcn_wmma_f32_16x16x4_f32(false, a0, false, b0, (short)0, acc, false, false);
                acc = __builtin_amdgcn_wmma_f32_16x16x4_f32(false, a1, false, b1, (short)0, acc, false, false);
                acc = __builtin_amdgcn_wmma_f32_16x16x4_f32(false, a2, false, b2, (short)0, acc, false, false);
                acc = __builtin_amdgcn_wmma_f32_16x16x4_f32(false, a3, false, b3, (short)0, acc, false, false);
            }
        }

        // ---- epilogue: D layout = lane L, VGPR v holds Y[M = v + 8*hi][N = lo]
        const float bv = bias[rs * BB + lo];
        const int ocol = rs * BB + lo;
        #pragma unroll
        for (int v = 0; v < 8; ++v) {
            const int m = v + 8 * hi;
            out[(size_t)(n0 + m) * XSTRIDE + ocol] = acc[v] + bv;
        }
    }
}

extern "C" void kernel_launch(void* const* d_in, const int* in_sizes, int n_in,
                              void* d_out, int out_size, void* d_ws, size_t ws_size,
                              hipStream_t stream) {
    const float* x      = (const float*)d_in[0];
    const float* values = (const float*)d_in[1];
    const float* bias   = (const float*)d_in[2];
    const int*   col    = (const int*)d_in[3];
    float* out          = (float*)d_out;

    const int blocks = NN / BB;  // 512 workgroups, one per 16-row n-tile
    blockell_lds_wmma_f32<<<blocks, 1024, LDS_BYTES, stream>>>(x, values, bias, col, out);
}